// MambaMIL_87814901334306
// MI455X (gfx1250) — compile-verified
//
#include <hip/hip_runtime.h>
#include <hip/hip_bf16.h>
#include <math.h>

typedef __attribute__((ext_vector_type(2))) float v2f;
typedef __attribute__((ext_vector_type(8))) float v8f;

#define NTOK      20000
#define PATH_DIM  1024
#define FEAT_DIM  64
#define D_INNER   128
#define D_STATE   16
#define DT_RANK   4
#define CHUNK     128
#define NCHUNK    157          // ceil(20000/128)

// ---------------------------------------------------------------------------
// Generic WMMA f32 GEMM:  C[m,n] = act( sum_k A[m,k] * W[n,k] )
// Block = 256 threads = 8 waves. Tile: BM=128 (16 rows/wave), BN=64, BK=32.
// Double-buffered LDS fed by GLOBAL_LOAD_ASYNC_TO_LDS_B128 (ASYNCcnt),
// software-pipelined: tile i+1 loads overlap tile i WMMA compute.
// MODE 0: plain   MODE 1: relu(x + bias[n])   MODE 2: x + residual[m,n]
// ---------------------------------------------------------------------------
#define TSTR 36   // padded row stride (floats): 16B-aligned b128 stores,
                  // conflict-free b64 fragment reads (36l mod 64 distinct, +2 disjoint)

__device__ __forceinline__ void async_b128(const float* g, const float* lds)
{
    asm volatile("global_load_async_to_lds_b128 %0, %1, off"
                 :
                 : "v"((unsigned)(unsigned long long)lds),
                   "v"((unsigned long long)g)
                 : "memory");
}

template <int MODE>
__global__ __launch_bounds__(256) void wmma_gemm_kernel(
    const float* __restrict__ A, const float* __restrict__ W,
    const float* __restrict__ aux, float* __restrict__ C,
    int M, int K, int Ntot)
{
    __shared__ __align__(16) float As[2][128 * TSTR];
    __shared__ __align__(16) float Bs[2][64 * TSTR];

    const int t    = threadIdx.x;
    const int warp = t >> 5;
    const int lane = t & 31;
    const int l16  = lane & 15;
    const int hi   = lane >> 4;
    const int bm   = blockIdx.x * 128;
    const int bn   = blockIdx.y * 64;

    // per-thread staging coordinates (6 async b128 per tile: 4 for A, 2 for B)
    const int ac4 = (t & 7) * 4;       // A: k-offset within tile
    const int arb = t >> 3;            // A: row base (0..31), 4 rows stride 32
    const v8f zero8 = {0.f, 0.f, 0.f, 0.f, 0.f, 0.f, 0.f, 0.f};
    v8f acc[4] = {zero8, zero8, zero8, zero8};

    // ---- issue one tile's async loads into buffer b at k-offset ko ----
    auto issue_tile = [&](int b, int ko) {
#pragma unroll
        for (int p = 0; p < 4; ++p) {
            const int row = p * 32 + arb;
            int gr = bm + row;
            gr = (gr < M) ? gr : (M - 1);      // clamp: garbage rows never stored
            async_b128(A + (size_t)gr * K + ko + ac4, &As[b][row * TSTR + ac4]);
        }
#pragma unroll
        for (int p = 0; p < 2; ++p) {
            const int idx = p * 256 + t;       // 0..511
            const int n   = idx >> 3;          // 0..63
            const int k4  = (idx & 7) * 4;     // 0..28
            async_b128(W + (size_t)(bn + n) * K + ko + k4, &Bs[b][n * TSTR + k4]);
        }
    };

    issue_tile(0, 0);                          // prologue: tile 0 in flight

    const int ntiles = K >> 5;
    for (int it = 0; it < ntiles; ++it) {
        const int cur = it & 1;
        // pipeline: start tile it+1 (clamped on last iter; lands in unused buffer)
        const int kon = (it + 1 < ntiles) ? (it + 1) * 32 : (ntiles - 1) * 32;
        issue_tile(cur ^ 1, kon);
        // wait for tile `it` only: 12 outstanding, in-order done => <=6 is ours
        asm volatile("s_wait_asynccnt 0x6" ::: "memory");
        __syncthreads();

        // ---- 8 k-steps of V_WMMA_F32_16X16X4_F32 over 4 N-subtiles ----
#pragma unroll
        for (int kk = 0; kk < 32; kk += 4) {
            // A frag (ISA 7.12.2, f32 A 16x4): lanes 0-15 K={kk,kk+1},
            // lanes 16-31 K={kk+2,kk+3} -> one b64 LDS read.
            v2f af = *reinterpret_cast<const v2f*>(
                &As[cur][(warp * 16 + l16) * TSTR + kk + 2 * hi]);
#pragma unroll
            for (int nt = 0; nt < 4; ++nt) {
                // B frag mirrors A: lane carries N=l16, components K -> one b64 read
                v2f bf = *reinterpret_cast<const v2f*>(
                    &Bs[cur][(nt * 16 + l16) * TSTR + kk + 2 * hi]);
                acc[nt] = __builtin_amdgcn_wmma_f32_16x16x4_f32(
                    false, af, false, bf, (short)0, acc[nt], false, false);
            }
        }
        __syncthreads();   // protect cur buffer from next iteration's issue
    }

    // ---- epilogue: C/D layout = VGPR r -> M = r + 8*hi, N = lane%16 ----
#pragma unroll
    for (int nt = 0; nt < 4; ++nt) {
        const int col = bn + nt * 16 + l16;
#pragma unroll
        for (int r = 0; r < 8; ++r) {
            const int row = bm + warp * 16 + r + 8 * hi;
            if (row < M) {
                float v = acc[nt][r];
                if (MODE == 1) v = fmaxf(v + aux[col], 0.0f);
                if (MODE == 2) v += aux[(size_t)row * Ntot + col];
                C[(size_t)row * Ntot + col] = v;
            }
        }
    }
}

// ---------------------------------------------------------------------------
// Depthwise causal conv (width 4) + bias + SiLU over the x-half of xz.
// ---------------------------------------------------------------------------
__global__ __launch_bounds__(256) void conv_silu_kernel(
    const float* __restrict__ xz, const float* __restrict__ cw,
    const float* __restrict__ cb, float* __restrict__ xc, int Ntok)
{
    const int t = threadIdx.x;
    const int c = t & 127;
    const int n = blockIdx.x * 2 + (t >> 7);
    if (n >= Ntok) return;
    float acc = cb[c];
#pragma unroll
    for (int j = 0; j < 4; ++j) {
        const int m = n - 3 + j;
        const float xv = (m >= 0) ? xz[(size_t)m * 256 + c] : 0.0f;
        acc = fmaf(cw[c * 4 + j], xv, acc);
    }
    const float sig = 1.0f / (1.0f + __expf(-acc));
    xc[(size_t)n * 128 + c] = acc * sig;
}

// ---------------------------------------------------------------------------
// Per-token x_proj (36 outs over K=128) + dt_proj (K=4) + softplus.
// ---------------------------------------------------------------------------
__global__ __launch_bounds__(128) void xproj_kernel(
    const float* __restrict__ xc, const float* __restrict__ xw,
    const float* __restrict__ dtw, const float* __restrict__ dtb,
    float* __restrict__ dt, float* __restrict__ Bm, float* __restrict__ Cm)
{
    __shared__ float xrow[128];
    __shared__ float xdbl[36];
    const int n = blockIdx.x;
    const int t = threadIdx.x;
    xrow[t] = xc[(size_t)n * 128 + t];
    __syncthreads();
    if (t < 36) {
        float s = 0.f;
        for (int k = 0; k < 128; ++k) s = fmaf(xrow[k], xw[t * 128 + k], s);
        xdbl[t] = s;
    }
    __syncthreads();
    float s = dtb[t];
#pragma unroll
    for (int r = 0; r < 4; ++r) s = fmaf(xdbl[r], dtw[t * 4 + r], s);
    const float sp = (s > 20.f) ? s : log1pf(__expf(s));   // softplus
    dt[(size_t)n * 128 + t] = sp;
    if (t < 16) {
        Bm[(size_t)n * 16 + t] = xdbl[4 + t];
        Cm[(size_t)n * 16 + t] = xdbl[20 + t];
    }
}

// ---------------------------------------------------------------------------
// Chunked parallel linear-recurrence scan, 3 phases.
// thread t of 256 owns channel c = t>>1, states sb..sb+7 (sb = (t&1)*8);
// flat pair index = t*8 + i.
// ---------------------------------------------------------------------------
__global__ __launch_bounds__(256) void scan_phase1(
    const float* __restrict__ dt, const float* __restrict__ x,
    const float* __restrict__ Bm, const float* __restrict__ A_log,
    float* __restrict__ P, float* __restrict__ Q, int Ntok)
{
    const int g = blockIdx.x, t = threadIdx.x;
    const int c = t >> 1, sb = (t & 1) * 8;
    float a[8], p[8], q[8];
#pragma unroll
    for (int i = 0; i < 8; ++i) {
        a[i] = -__expf(A_log[c * 16 + sb + i]);
        p[i] = 1.f;
        q[i] = 0.f;
    }
    const int n0 = g * CHUNK;
    const int n1 = min(n0 + CHUNK, Ntok);
    for (int n = n0; n < n1; ++n) {
        const float dtv = dt[(size_t)n * 128 + c];
        const float dx  = dtv * x[(size_t)n * 128 + c];
#pragma unroll
        for (int i = 0; i < 8; ++i) {
            const float e = __expf(dtv * a[i]);
            q[i] = fmaf(e, q[i], dx * Bm[(size_t)n * 16 + sb + i]);
            p[i] *= e;
        }
    }
#pragma unroll
    for (int i = 0; i < 8; ++i) {
        P[(size_t)g * 2048 + t * 8 + i] = p[i];
        Q[(size_t)g * 2048 + t * 8 + i] = q[i];
    }
}

__global__ __launch_bounds__(256) void scan_phase2(
    const float* __restrict__ P, const float* __restrict__ Q,
    float* __restrict__ Hin, int G)
{
    const int t = threadIdx.x;
    float h[8] = {0.f, 0.f, 0.f, 0.f, 0.f, 0.f, 0.f, 0.f};
    for (int g = 0; g < G; ++g) {
        const size_t base = (size_t)g * 2048 + t * 8;
#pragma unroll
        for (int i = 0; i < 8; ++i) {
            Hin[base + i] = h[i];
            h[i] = fmaf(P[base + i], h[i], Q[base + i]);
        }
    }
}

__global__ __launch_bounds__(256) void scan_phase3(
    const float* __restrict__ dt, const float* __restrict__ x,
    const float* __restrict__ Bm, const float* __restrict__ Cm,
    const float* __restrict__ A_log, const float* __restrict__ Dskip,
    const float* __restrict__ xz, const float* __restrict__ Hin,
    float* __restrict__ y, int Ntok)
{
    const int g = blockIdx.x, t = threadIdx.x;
    const int c = t >> 1, sb = (t & 1) * 8;
    float a[8], h[8];
#pragma unroll
    for (int i = 0; i < 8; ++i) {
        a[i] = -__expf(A_log[c * 16 + sb + i]);
        h[i] = Hin[(size_t)g * 2048 + t * 8 + i];
    }
    const float dsk = Dskip[c];
    const int n0 = g * CHUNK;
    const int n1 = min(n0 + CHUNK, Ntok);
    for (int n = n0; n < n1; ++n) {
        const float dtv = dt[(size_t)n * 128 + c];
        const float xv  = x[(size_t)n * 128 + c];
        const float dx  = dtv * xv;
        float partial = 0.f;
#pragma unroll
        for (int i = 0; i < 8; ++i) {
            const float e = __expf(dtv * a[i]);
            h[i] = fmaf(e, h[i], dx * Bm[(size_t)n * 16 + sb + i]);
            partial = fmaf(h[i], Cm[(size_t)n * 16 + sb + i], partial);
        }
        const float tot = partial + __shfl_xor(partial, 1, 32);
        if ((t & 1) == 0) {
            const float zv  = xz[(size_t)n * 256 + 128 + c];
            const float sig = 1.0f / (1.0f + __expf(-zv));
            y[(size_t)n * 128 + c] = (tot + dsk * xv) * zv * sig;
        }
    }
}

// ---------------------------------------------------------------------------
// Attention MLP logits:  a[n] = tanh(feat@w1^T + b1) @ w2^T + b2
// ---------------------------------------------------------------------------
__global__ __launch_bounds__(256) void attn_kernel(
    const float* __restrict__ feat, const float* __restrict__ w1,
    const float* __restrict__ b1, const float* __restrict__ w2,
    const float* __restrict__ b2, float* __restrict__ abuf, int Ntok)
{
    const int n = blockIdx.x * 256 + threadIdx.x;
    if (n >= Ntok) return;
    float h[16];
#pragma unroll
    for (int j = 0; j < 16; ++j) h[j] = b1[j];
    for (int k = 0; k < 64; ++k) {
        const float f = feat[(size_t)n * 64 + k];
#pragma unroll
        for (int j = 0; j < 16; ++j) h[j] = fmaf(f, w1[j * 64 + k], h[j]);
    }
    float a = b2[0];
#pragma unroll
    for (int j = 0; j < 16; ++j) a = fmaf(tanhf(h[j]), w2[j], a);
    abuf[n] = a;
}

// ---------------------------------------------------------------------------
// Single-block softmax over N logits -> out (d_out+64).
// ---------------------------------------------------------------------------
__global__ __launch_bounds__(1024) void softmax_kernel(
    const float* __restrict__ a, float* __restrict__ out, int n)
{
    __shared__ float red[1024];
    const int t = threadIdx.x;
    float m = -3.402823e38f;
    for (int i = t; i < n; i += 1024) m = fmaxf(m, a[i]);
    red[t] = m;
    __syncthreads();
    for (int s = 512; s > 0; s >>= 1) {
        if (t < s) red[t] = fmaxf(red[t], red[t + s]);
        __syncthreads();
    }
    const float mx = red[0];
    __syncthreads();
    float sum = 0.f;
    for (int i = t; i < n; i += 1024) sum += __expf(a[i] - mx);
    red[t] = sum;
    __syncthreads();
    for (int s = 512; s > 0; s >>= 1) {
        if (t < s) red[t] += red[t + s];
        __syncthreads();
    }
    const float inv = 1.0f / red[0];
    for (int i = t; i < n; i += 1024) out[i] = __expf(a[i] - mx) * inv;
}

// ---------------------------------------------------------------------------
// M[f] = sum_n A_soft[n] * feature[n,f]   (single block)
// ---------------------------------------------------------------------------
__global__ __launch_bounds__(256) void pool_kernel(
    const float* __restrict__ asoft, const float* __restrict__ feature,
    float* __restrict__ Mout, int n)
{
    __shared__ float red[256];
    const int t = threadIdx.x;
    const int f = t & 63;
    const int sl = t >> 6;
    float acc = 0.f;
    for (int i = sl; i < n; i += 4)
        acc = fmaf(asoft[i], feature[(size_t)i * 64 + f], acc);
    red[t] = acc;
    __syncthreads();
    if (t < 64) Mout[t] = red[t] + red[t + 64] + red[t + 128] + red[t + 192];
}

// ---------------------------------------------------------------------------
extern "C" void kernel_launch(void* const* d_in, const int* in_sizes, int n_in,
                              void* d_out, int out_size, void* d_ws, size_t ws_size,
                              hipStream_t stream)
{
    (void)in_sizes; (void)n_in; (void)out_size; (void)ws_size;
    const float* x_path    = (const float*)d_in[0];
    const float* W1        = (const float*)d_in[1];
    const float* b1        = (const float*)d_in[2];
    const float* in_proj_w = (const float*)d_in[3];
    const float* conv_w    = (const float*)d_in[4];
    const float* conv_b    = (const float*)d_in[5];
    const float* x_proj_w  = (const float*)d_in[6];
    const float* dt_proj_w = (const float*)d_in[7];
    const float* dt_proj_b = (const float*)d_in[8];
    const float* A_log     = (const float*)d_in[9];
    const float* Dskip     = (const float*)d_in[10];
    const float* out_proj_w= (const float*)d_in[11];
    const float* attn_w1   = (const float*)d_in[12];
    const float* attn_b1   = (const float*)d_in[13];
    const float* attn_w2   = (const float*)d_in[14];
    const float* attn_b2   = (const float*)d_in[15];

    float* out = (float*)d_out;           // [ M(64) | A_soft(20000) ]
    float* ws  = (float*)d_ws;

    size_t off = 0;
    float* feature = ws + off; off += (size_t)NTOK * 64;
    float* xz      = ws + off; off += (size_t)NTOK * 256;
    float* xc      = ws + off; off += (size_t)NTOK * 128;
    float* dt      = ws + off; off += (size_t)NTOK * 128;
    float* Bm      = ws + off; off += (size_t)NTOK * 16;
    float* Cm      = ws + off; off += (size_t)NTOK * 16;
    float* P       = ws + off; off += (size_t)NCHUNK * 2048;
    float* Q       = ws + off; off += (size_t)NCHUNK * 2048;
    float* Hin     = ws + off; off += (size_t)NCHUNK * 2048;
    float* ybuf    = ws + off; off += (size_t)NTOK * 128;
    float* feat2   = ws + off; off += (size_t)NTOK * 64;
    float* abuf    = ws + off; off += (size_t)NTOK;

    const dim3 blk(256);
    const int mtiles = (NTOK + 127) / 128;   // 157

    // 1) feature = relu(x_path @ W1^T + b1)     [WMMA f32, async-LDS pipelined]
    wmma_gemm_kernel<1><<<dim3(mtiles, 1), blk, 0, stream>>>(
        x_path, W1, b1, feature, NTOK, PATH_DIM, FEAT_DIM);
    // 2) xz = feature @ in_proj_w^T             [WMMA f32]
    wmma_gemm_kernel<0><<<dim3(mtiles, 4), blk, 0, stream>>>(
        feature, in_proj_w, nullptr, xz, NTOK, FEAT_DIM, 2 * D_INNER);
    // 3) depthwise causal conv + SiLU
    conv_silu_kernel<<<dim3((NTOK + 1) / 2), blk, 0, stream>>>(
        xz, conv_w, conv_b, xc, NTOK);
    // 4) x_proj + dt_proj + softplus
    xproj_kernel<<<dim3(NTOK), dim3(128), 0, stream>>>(
        xc, x_proj_w, dt_proj_w, dt_proj_b, dt, Bm, Cm);
    // 5-7) chunked parallel selective scan + fused gating
    scan_phase1<<<dim3(NCHUNK), blk, 0, stream>>>(dt, xc, Bm, A_log, P, Q, NTOK);
    scan_phase2<<<dim3(1), blk, 0, stream>>>(P, Q, Hin, NCHUNK);
    scan_phase3<<<dim3(NCHUNK), blk, 0, stream>>>(
        dt, xc, Bm, Cm, A_log, Dskip, xz, Hin, ybuf, NTOK);
    // 8) feat2 = y @ out_proj_w^T + feature     [WMMA f32, residual epilogue]
    wmma_gemm_kernel<2><<<dim3(mtiles, 1), blk, 0, stream>>>(
        ybuf, out_proj_w, feature, feat2, NTOK, D_INNER, FEAT_DIM);
    // 9) attention logits
    attn_kernel<<<dim3((NTOK + 255) / 256), blk, 0, stream>>>(
        feat2, attn_w1, attn_b1, attn_w2, attn_b2, abuf, NTOK);
    // 10) softmax -> A_soft at out+64
    softmax_kernel<<<dim3(1), dim3(1024), 0, stream>>>(abuf, out + 64, NTOK);
    // 11) M = A_soft @ feature -> out[0..63]
    pool_kernel<<<dim3(1), blk, 0, stream>>>(out + 64, feature, out, NTOK);
}